// FlowRNN_69965017252729
// MI455X (gfx1250) — compile-verified
//
#include <hip/hip_runtime.h>
#include <stdint.h>
#include <stddef.h>

// ---------------------------------------------------------------------------
// FlowRNN (GRU with flow-delta residual input) for MI455X / gfx1250.
//
//  B=128, T=1024, D_IN=512, H=256, GRU_IN=768.
//
//  Phase 0: pack weights into WMMA B-fragment layout, zero h/res carry.
//  Phase 1: gxX[t][b][3H] = x_t . W_x^T + b_ih   (parallel, fp32 WMMA 16x16x4)
//  Phase 2: sequential scan, 8 independent workgroups (one 16-row batch tile
//           each), per step a 16x(4H')x512 fp32 WMMA GEMM:
//             cols [0,256)   r-gate,  K=512 over [res|h]
//             cols [256,512) z-gate,  K=512 over [res|h]
//             cols [512,768) n-res ,  K=256 over res only
//             cols [768,1024)n-h   ,  K=256 over h  only
//           then gates: r=s(gx_r+c_r+bhh_r), z=..., n=tanh(gx_n+c_nres+
//           r*(c_nh+bhh_n)), h' = n + z*(h-n), res' = h'-h.
// ---------------------------------------------------------------------------

typedef float v2f __attribute__((ext_vector_type(2)));
typedef float v8f __attribute__((ext_vector_type(8)));

#define B_    128
#define T_    1024
#define DIN   512
#define H_    256
#define K3H   768
#define ASTR  516   // LDS A stride (floats): 4*m banks -> conflict-free b64

// workspace layout in floats
#define PWX_OFF    0          // 48 tiles * 128 ksteps * 64 = 393216
#define PWREC_OFF  393216     // 64 tiles * 128 ksteps * 64 = 524288
#define CARRYH_OFF 917504     // 128*256
#define CARRYR_OFF 950272     // 128*256
#define GX_OFF     1048576    // chunk buffer: TC*128*768

static __device__ inline v8f wmma4(v2f a, v2f b, v8f c) {
  // D = A(16x4 f32) x B(4x16 f32) + C(16x16 f32)   -> v_wmma_f32_16x16x4_f32
  return __builtin_amdgcn_wmma_f32_16x16x4_f32(false, a, false, b,
                                               (short)0, c, false, false);
}
static __device__ inline float fexp(float x) {
  return __builtin_amdgcn_exp2f(x * 1.4426950408889634f);
}
static __device__ inline float fsig(float x) {
  return __builtin_amdgcn_rcpf(1.0f + fexp(-x));
}
static __device__ inline float ftanh(float x) {
  float e = fexp(-2.0f * x);
  return (1.0f - e) * __builtin_amdgcn_rcpf(1.0f + e);
}

// ---------------------------------------------------------------------------
// Phase 0: pack W into per-lane B-fragment layout; zero carry buffers.
// B fragment for tile jt, kstep s: flat[((jt*128+s)*64) + lane*2 + e] =
//   W[n = 16*jt' + lane%16][k = 4*s + 2*(lane/16) + e]
// ---------------------------------------------------------------------------
__global__ __launch_bounds__(256) void pack_kernel(
    const float* __restrict__ w_ih, const float* __restrict__ w_hh,
    float* __restrict__ ws) {
  int idx = blockIdx.x * 256 + threadIdx.x;        // 0 .. 917503
  if (idx < 2 * B_ * H_) ws[CARRYH_OFF + idx] = 0.0f;  // zero h & res carry

  if (idx < 393216) {                              // pw_x: x-part of w_ih
    int e = idx & 1, L = (idx >> 1) & 31, s = (idx >> 6) & 127, j = idx >> 13;
    int n = 16 * j + (L & 15);
    int k = 4 * s + 2 * (L >> 4) + e;              // 0..511
    ws[PWX_OFF + idx] = w_ih[(size_t)n * K3H + 256 + k];
  } else {                                         // pw_rec
    int i2 = idx - 393216;                         // 0 .. 524287
    int e = i2 & 1, L = (i2 >> 1) & 31, s = (i2 >> 6) & 127, jt = i2 >> 13;
    int c16 = L & 15, hh = L >> 4;
    int k = 4 * s + 2 * hh + e;
    float v = 0.0f;
    if (jt < 32) {                                 // r (jt<16) / z rows, K=512
      int n = (jt < 16 ? 16 * jt : 256 + 16 * (jt - 16)) + c16;
      v = (k < 256) ? w_ih[(size_t)n * K3H + k]    // res columns of w_ih
                    : w_hh[(size_t)n * H_ + (k - 256)];
    } else if (jt < 48) {                          // n-gate, res-only, K=256
      int n = 512 + 16 * (jt - 32) + c16;
      if (s < 64) v = w_ih[(size_t)n * K3H + k];
    } else {                                       // n-gate, h-only, K=256
      int n = 512 + 16 * (jt - 48) + c16;
      if (s < 64) v = w_hh[(size_t)n * H_ + k];
    }
    ws[PWREC_OFF + i2] = v;
  }
}

// ---------------------------------------------------------------------------
// Phase 1: gxX[tl][b][n] = x[b][t] . W_x^T + b_ih.  One WG = 32 batch rows at
// one t; A staged via async global->LDS; 8 waves * 6 N-tiles * 2 M-tiles.
// ---------------------------------------------------------------------------
__global__ __launch_bounds__(256) void gemm_x_kernel(
    const float* __restrict__ x, const float* __restrict__ pwx,
    const float* __restrict__ b_ih, float* __restrict__ gx, int t0, int tc) {
  extern __shared__ float As[];                    // 32 x ASTR floats
  const int tid = threadIdx.x, lane = tid & 31, wave = tid >> 5;
  const int tl = blockIdx.x >> 2;
  const int t  = t0 + tl;
  const int b0 = (blockIdx.x & 3) * 32;
  const int ma = lane & 15, hh = lane >> 4;

  // ---- stage A (32 rows x 512 f32) with async tensor-path copies ----------
  unsigned lds0 = (unsigned)(size_t)(&As[0]);      // LDS byte offset
  for (int i = 0; i < 4; ++i) {
    int r = wave + 8 * i;
    const char* src = (const char*)(x + ((size_t)(b0 + r) * T_ + t) * DIN);
    unsigned dst = lds0 + (unsigned)(r * ASTR * 4);
#pragma unroll
    for (int p = 0; p < 4; ++p) {
      uint64_t ga = (uint64_t)(src + p * 512 + lane * 16);
      unsigned  la = dst + p * 512 + lane * 16;
      asm volatile("global_load_async_to_lds_b128 %0, %1, off"
                   :: "v"(la), "v"(ga) : "memory");
    }
  }
  asm volatile("s_wait_asynccnt 0" ::: "memory");
  __syncthreads();

  // ---- GEMM: C[6 N-tiles][2 M-tiles], K = 512 -----------------------------
  const float* pw = pwx + 2 * lane;                // per-lane fragment base
  v8f C[6][2];
#pragma unroll
  for (int nt = 0; nt < 6; ++nt) { C[nt][0] = (v8f){}; C[nt][1] = (v8f){}; }

#pragma unroll 4
  for (int s = 0; s < 128; ++s) {
    v2f a0 = *(const v2f*)&As[ma * ASTR + 4 * s + 2 * hh];
    v2f a1 = *(const v2f*)&As[(ma + 16) * ASTR + 4 * s + 2 * hh];
#pragma unroll
    for (int nt = 0; nt < 6; ++nt) {
      int jt = 6 * wave + nt;
      v2f b = *(const v2f*)&pw[(jt * 128 + s) << 6];
      C[nt][0] = wmma4(a0, b, C[nt][0]);
      C[nt][1] = wmma4(a1, b, C[nt][1]);
    }
  }

  // ---- epilogue: add b_ih, store gx[tl*128 + b][col] ----------------------
#pragma unroll
  for (int nt = 0; nt < 6; ++nt) {
    int jt = 6 * wave + nt;
    int col = 16 * jt + ma;
    float bias = b_ih[col];
#pragma unroll
    for (int mt = 0; mt < 2; ++mt)
#pragma unroll
      for (int v = 0; v < 8; ++v) {
        int m = mt * 16 + v + 8 * hh;
        gx[(size_t)(tl * 128 + b0 + m) * K3H + col] = C[nt][mt][v] + bias;
      }
  }
}

// ---------------------------------------------------------------------------
// Phase 2: sequential scan.  8 WGs, one 16-row batch tile each, tc steps.
// ---------------------------------------------------------------------------
__global__ __launch_bounds__(256) void scan_kernel(
    const float* __restrict__ gx, const float* __restrict__ pwrec,
    const float* __restrict__ b_hh, float* __restrict__ carry_h,
    float* __restrict__ carry_r, float* __restrict__ out,
    float* __restrict__ resout, int t0, int tc) {
  __shared__ float As[16 * ASTR];                  // [m][0:256)=res, [256:512)=h
  const int tid = threadIdx.x, lane = tid & 31, wave = tid >> 5;
  const int b0 = blockIdx.x * 16;
  const int ma = lane & 15, hh = lane >> 4;

  // stage A from carry buffers (zeroed by pack_kernel for t0 == 0)
  for (int i = tid; i < 16 * 256; i += 256) {
    int m = i >> 8, k = i & 255;
    As[m * ASTR + k]       = carry_r[(b0 + m) * H_ + k];
    As[m * ASTR + 256 + k] = carry_h[(b0 + m) * H_ + k];
  }

  int hc[2];
  float bhr[2], bhz[2], bhn[2];
  float hp[2][8], ri[2][8];                        // h_prev, res_in (C layout)
#pragma unroll
  for (int u = 0; u < 2; ++u) {
    hc[u] = 32 * wave + 16 * u + ma;
    bhr[u] = b_hh[hc[u]];
    bhz[u] = b_hh[256 + hc[u]];
    bhn[u] = b_hh[512 + hc[u]];
#pragma unroll
    for (int v = 0; v < 8; ++v) {
      int m = v + 8 * hh;
      hp[u][v] = carry_h[(b0 + m) * H_ + hc[u]];
      ri[u][v] = carry_r[(b0 + m) * H_ + hc[u]];
    }
  }
  __syncthreads();

  const float* pw = pwrec + 2 * lane;
  const int jr = 2 * wave, jz = 16 + 2 * wave, jnr = 32 + 2 * wave,
            jnh = 48 + 2 * wave;

  for (int tl = 0; tl < tc; ++tl) {
    const int t = t0 + tl;

    // precomputed input-GEMM terms for this step (latency hidden by GEMM)
    float gr[2][8], gz[2][8], gn[2][8];
#pragma unroll
    for (int u = 0; u < 2; ++u)
#pragma unroll
      for (int v = 0; v < 8; ++v) {
        const float* gm = gx + (size_t)(tl * 128 + b0 + v + 8 * hh) * K3H;
        gr[u][v] = gm[hc[u]];
        gz[u][v] = gm[256 + hc[u]];
        gn[u][v] = gm[512 + hc[u]];
        if (tl + 1 < tc)                           // warm next batch row's lines, same step
          __builtin_prefetch(gm + K3H + hc[u], 0, 0);
      }

    // ---- recurrent GEMM: [res|h](16x512) x Wrec -> 8 accumulators ---------
    v8f cr[2], cz[2], cnr[2], cnh[2];
#pragma unroll
    for (int u = 0; u < 2; ++u) {
      cr[u] = (v8f){}; cz[u] = (v8f){}; cnr[u] = (v8f){}; cnh[u] = (v8f){};
    }
#pragma unroll 4
    for (int s = 0; s < 64; ++s) {
      v2f aR = *(const v2f*)&As[ma * ASTR + 4 * s + 2 * hh];        // res part
      v2f aH = *(const v2f*)&As[ma * ASTR + 256 + 4 * s + 2 * hh];  // h part
#pragma unroll
      for (int u = 0; u < 2; ++u) {   // aR-based (independent accums first)
        v2f b0r = *(const v2f*)&pw[((jr + u) * 128 + s) << 6];
        v2f b0z = *(const v2f*)&pw[((jz + u) * 128 + s) << 6];
        v2f bnr = *(const v2f*)&pw[((jnr + u) * 128 + s) << 6];
        cr[u]  = wmma4(aR, b0r, cr[u]);
        cz[u]  = wmma4(aR, b0z, cz[u]);
        cnr[u] = wmma4(aR, bnr, cnr[u]);
      }
#pragma unroll
      for (int u = 0; u < 2; ++u) {   // aH-based
        v2f b1r = *(const v2f*)&pw[((jr + u) * 128 + s + 64) << 6];
        v2f b1z = *(const v2f*)&pw[((jz + u) * 128 + s + 64) << 6];
        v2f bnh = *(const v2f*)&pw[((jnh + u) * 128 + s) << 6];
        cr[u]  = wmma4(aH, b1r, cr[u]);
        cz[u]  = wmma4(aH, b1z, cz[u]);
        cnh[u] = wmma4(aH, bnh, cnh[u]);
      }
    }

    // ---- gates + state update ---------------------------------------------
#pragma unroll
    for (int u = 0; u < 2; ++u)
#pragma unroll
      for (int v = 0; v < 8; ++v) {
        float r  = fsig(gr[u][v] + cr[u][v] + bhr[u]);
        float z  = fsig(gz[u][v] + cz[u][v] + bhz[u]);
        float nn = ftanh(gn[u][v] + cnr[u][v] + r * (cnh[u][v] + bhn[u]));
        float hn = nn + z * (hp[u][v] - nn);
        size_t ob = ((size_t)(b0 + v + 8 * hh) * T_ + t) * H_ + hc[u];
        out[ob]    = hn;
        resout[ob] = ri[u][v];                     // residual logged pre-update
        ri[u][v] = hn - hp[u][v];
        hp[u][v] = hn;
      }

    __syncthreads();                               // all waves done reading As
#pragma unroll
    for (int u = 0; u < 2; ++u)
#pragma unroll
      for (int v = 0; v < 8; ++v) {
        int m = v + 8 * hh;
        As[m * ASTR + hc[u]]       = ri[u][v];     // next-step res input
        As[m * ASTR + 256 + hc[u]] = hp[u][v];     // next-step h input
      }
    __syncthreads();
  }

  // persist carry for next chunk
#pragma unroll
  for (int u = 0; u < 2; ++u)
#pragma unroll
    for (int v = 0; v < 8; ++v) {
      int m = v + 8 * hh;
      carry_h[(b0 + m) * H_ + hc[u]] = hp[u][v];
      carry_r[(b0 + m) * H_ + hc[u]] = ri[u][v];
    }
}

// ---------------------------------------------------------------------------
extern "C" void kernel_launch(void* const* d_in, const int* in_sizes, int n_in,
                              void* d_out, int out_size, void* d_ws,
                              size_t ws_size, hipStream_t stream) {
  const float* x    = (const float*)d_in[0];
  const float* w_ih = (const float*)d_in[1];
  const float* w_hh = (const float*)d_in[2];
  const float* b_ih = (const float*)d_in[3];
  const float* b_hh = (const float*)d_in[4];

  float* ws     = (float*)d_ws;
  float* out    = (float*)d_out;
  float* resout = out + (size_t)B_ * T_ * H_;
  float* pwx    = ws + PWX_OFF;
  float* pwrec  = ws + PWREC_OFF;
  float* ch     = ws + CARRYH_OFF;
  float* crr    = ws + CARRYR_OFF;
  float* gxbuf  = ws + GX_OFF;

  // T-chunk sized to fit workspace (gx chunk = TC*128*768 floats)
  size_t avail = (ws_size / 4 > (size_t)GX_OFF) ? ws_size / 4 - GX_OFF : 0;
  int TC = (int)(avail / (128 * 768));
  if (TC > T_) TC = T_;
  TC &= ~31;
  if (TC < 32) TC = 32;

  pack_kernel<<<3584, 256, 0, stream>>>(w_ih, w_hh, ws);
  for (int t0 = 0; t0 < T_; t0 += TC) {
    int tc = (T_ - t0 < TC) ? (T_ - t0) : TC;
    gemm_x_kernel<<<tc * 4, 256, 32 * ASTR * 4, stream>>>(x, pwx, b_ih, gxbuf,
                                                          t0, tc);
    scan_kernel<<<8, 256, 0, stream>>>(gxbuf, pwrec, b_hh, ch, crr, out,
                                       resout, t0, tc);
  }
}